// AUX_OVER_IVA_ONLINE_12232066859625
// MI455X (gfx1250) — compile-verified
//
#include <hip/hip_runtime.h>
#include <math.h>

// ---------------- problem constants (match reference) ----------------
#define M_MICS   6
#define K_SRC    2
#define T_FRAMES 1000
#define F_BINS   1025
#define ALPHA_F  0.96f
#define EPS_R_F  1e-10f
#define REG_F    1e-6f

// ---------------- launch geometry ----------------
#define NWG   9          // persistent workgroups (fits in one 16-WG cluster)
#define BLOCK 128        // 4 wave32 per WG; NWG*BLOCK = 1152 >= F_BINS
#define NWAVE (BLOCK/32)

// ---------------- workspace layout (floats) ----------------
// [0]=barrier arrive cnt (u32)  [1]=barrier generation (u32)
// [2 .. 2+2*NWG*K_SRC) = ping-pong per-WG energy partials (buf = t&1)
#define WS_HDR 64
#define WS_C   WS_HDR                       // C state: F_BINS * 36 cplx (72 f)
#define WS_V   (WS_C + F_BINS*72)           // V state: 2 * F_BINS * 72 f
// total floats = WS_V + 2*F_BINS*72  (~886 KB; resident in 192MB L2)

typedef float v2f __attribute__((ext_vector_type(2)));   // A/B operand of wmma f32 16x16x4
typedef float v8f __attribute__((ext_vector_type(8)));   // C/D operand of wmma f32 16x16x4

struct cx { float r, i; };
__device__ __forceinline__ cx cmul(cx a, cx b){ return { a.r*b.r - a.i*b.i, a.r*b.i + a.i*b.r }; }
__device__ __forceinline__ cx cmac(cx acc, cx a, cx b){            // acc + a*b
  acc.r = fmaf(a.r, b.r, fmaf(-a.i, b.i, acc.r));
  acc.i = fmaf(a.r, b.i, fmaf( a.i, b.r, acc.i));
  return acc;
}
__device__ __forceinline__ cx cmsub(cx acc, cx a, cx b){           // acc - a*b
  acc.r = fmaf(-a.r, b.r, fmaf( a.i, b.i, acc.r));
  acc.i = fmaf(-a.r, b.i, fmaf(-a.i, b.r, acc.i));
  return acc;
}
__device__ __forceinline__ cx cinv(cx a){
  float s = 1.0f / fmaf(a.r, a.r, a.i*a.i);
  return { a.r*s, -a.i*s };
}

// 6x6 complex matrix as 18 float4 -> all global state traffic is b128.
struct mat6 { float4 q[18]; };
__device__ __forceinline__ cx m_get(const mat6& m, int i, int j){
  const float* f = (const float*)m.q;
  return { f[(i*6+j)*2], f[(i*6+j)*2 + 1] };
}
__device__ __forceinline__ void m_set(mat6& m, int i, int j, cx v){
  float* f = (float*)m.q;
  f[(i*6+j)*2] = v.r; f[(i*6+j)*2 + 1] = v.i;
}
__device__ __forceinline__ void m_load(mat6& m, const float* p){
  const float4* p4 = (const float4*)p;
  #pragma unroll
  for (int q = 0; q < 18; ++q) m.q[q] = p4[q];      // global_load_b128
}
__device__ __forceinline__ void m_store(const mat6& m, float* p){
  float4* p4 = (float4*)p;
  #pragma unroll
  for (int q = 0; q < 18; ++q) p4[q] = m.q[q];      // global_store_b128
}

// CDNA5 async gather: one complex sample (b64) memory -> this lane's LDS slot.
// Uses the ASYNCcnt path (GLOBAL_LOAD_ASYNC_TO_LDS_B64, CDNA5 ISA).
__device__ __forceinline__ void async_cplx_to_lds(const float2* g, float2* lds_slot){
  unsigned           lo = (unsigned)(uintptr_t)lds_slot;  // low 32b of flat = LDS offset
  unsigned long long ga = (unsigned long long)(uintptr_t)g;
  asm volatile("global_load_async_to_lds_b64 %0, %1, off" :: "v"(lo), "v"(ga) : "memory");
}
__device__ __forceinline__ void wait_async(){
  asm volatile("s_wait_asynccnt 0" ::: "memory");
}

// Device-scope barrier across the persistent grid, executed 1000x on the
// critical path -> minimal agent-scope fences, not full __threadfence().
// Fast path: single-cluster dispatch -> hardware cluster barrier.
// Fallback: sense-reversing atomic barrier in L2.
__device__ __forceinline__ void grid_barrier(unsigned* cnt, unsigned* gen, bool in_cluster){
  __syncthreads();
  __builtin_amdgcn_fence(__ATOMIC_RELEASE, "agent");
  if (in_cluster) {
    __builtin_amdgcn_s_cluster_barrier();
  } else if (threadIdx.x == 0) {
    unsigned g = __hip_atomic_load(gen, __ATOMIC_RELAXED, __HIP_MEMORY_SCOPE_AGENT);
    unsigned a = __hip_atomic_fetch_add(cnt, 1u, __ATOMIC_ACQ_REL, __HIP_MEMORY_SCOPE_AGENT);
    if (a == gridDim.x - 1) {
      __hip_atomic_store(cnt, 0u, __ATOMIC_RELAXED, __HIP_MEMORY_SCOPE_AGENT);
      __hip_atomic_fetch_add(gen, 1u, __ATOMIC_ACQ_REL, __HIP_MEMORY_SCOPE_AGENT);
    } else {
      while (__hip_atomic_load(gen, __ATOMIC_ACQUIRE, __HIP_MEMORY_SCOPE_AGENT) == g)
        __builtin_amdgcn_s_sleep(1);
    }
  }
  __builtin_amdgcn_fence(__ATOMIC_ACQUIRE, "agent");
  __syncthreads();
}

// ---------------- init: C0 = V0 = I per bin, zero barrier/partials ----------------
__global__ void oiva_init(float* __restrict__ ws){
  int i = blockIdx.x * blockDim.x + threadIdx.x;
  if (i < WS_HDR) ws[i] = 0.0f;
  if (i < F_BINS){
    float4* cp = (float4*)(ws + WS_C) + (size_t)i*18;
    float4* v0 = (float4*)(ws + WS_V) + (size_t)i*18;
    float4* v1 = (float4*)(ws + WS_V) + (size_t)F_BINS*18 + (size_t)i*18;
    #pragma unroll
    for (int q = 0; q < 18; ++q){
      // float4 covers elements (2q, 2q+1) of the row-major 6x6 cplx matrix
      int e0 = 2*q, e1 = 2*q + 1;
      float4 v = make_float4((e0/6 == e0%6) ? 1.0f : 0.0f, 0.0f,
                             (e1/6 == e1%6) ? 1.0f : 0.0f, 0.0f);
      cp[q] = v; v0[q] = v; v1[q] = v;
    }
  }
}

// ---------------- persistent online OverIVA ----------------
__global__ __launch_bounds__(BLOCK, 1)
void oiva_persist(const float2* __restrict__ X,   // (M, T, F) complex as float2
                  float2* __restrict__ out,       // (K, T, F) complex as float2
                  float* __restrict__ ws)
{
  __shared__ float2 sx[2][BLOCK * M_MICS];        // double-buffered x_t gather
  __shared__ float  sred[K_SRC][NWAVE];
  __shared__ float  sphi[K_SRC];                  // leader-broadcast contrast weights

  unsigned* cnt = (unsigned*)ws;
  unsigned* gen = ((unsigned*)ws) + 1;
  float* part = ws + 2;                           // [2][NWG][K_SRC] ping-pong
  float* Cst  = ws + WS_C;
  float* Vst  = ws + WS_V;

  const int  tid  = threadIdx.x;
  const int  lane = tid & 31, wave = tid >> 5;
  const int  f    = blockIdx.x * BLOCK + tid;     // one frequency bin per lane
  const bool act  = (f < F_BINS);
  const bool in_cluster = (__builtin_amdgcn_cluster_id_x() != 0);

  // B operand for the reduction WMMA (constant all steps): column selectors.
  // B[0][0]=B[2][0]=1 (sums e0 into column 0), B[1][1]=B[3][1]=1 (e1 -> col 1).
  // 4x16 B layout: V0 = rows K=0 (lanes 0-15) / K=1 (lanes 16-31); V1 = K=2/K=3.
  v2f Bsel;
  {
    float b = (lane == 0 || lane == 17) ? 1.0f : 0.0f;
    Bsel.x = b; Bsel.y = b;
  }

  // W_hat0 = eye(M), lower-right (M-K) block negated
  cx W[M_MICS][M_MICS];
  #pragma unroll
  for (int i2 = 0; i2 < M_MICS; ++i2)
    #pragma unroll
    for (int j = 0; j < M_MICS; ++j)
      W[i2][j] = { (i2==j) ? ((i2<K_SRC)?1.0f:-1.0f) : 0.0f, 0.0f };

  float2* slot0 = &sx[0][tid*M_MICS];
  float2* slot1 = &sx[1][tid*M_MICS];

  float* cp  = Cst + (size_t)(act ? f : 0) * 72;
  float* vp0 = Vst + (size_t)(act ? f : 0) * 72;
  float* vp1 = Vst + (size_t)F_BINS*72 + (size_t)(act ? f : 0) * 72;

  if (act) {                                       // prefetch x_0
    #pragma unroll
    for (int m = 0; m < M_MICS; ++m)
      async_cplx_to_lds(X + ((size_t)m*T_FRAMES + 0)*F_BINS + f, slot0 + m);
  }

  for (int t = 0; t < T_FRAMES; ++t){
    float2* cur = (t & 1) ? slot1 : slot0;
    float2* nxt = (t & 1) ? slot0 : slot1;
    float*  pbuf = part + (size_t)(t & 1) * (NWG * K_SRC);

    wait_async();                                  // x_t resident in LDS

    cx x[M_MICS];
    #pragma unroll
    for (int m = 0; m < M_MICS; ++m){
      float2 v = act ? cur[m] : make_float2(0.f, 0.f);
      x[m] = { v.x, v.y };
    }

    // x_t consumed -> immediately overlap the x_{t+1} gather with everything below
    if (act && (t+1) < T_FRAMES){
      #pragma unroll
      for (int m = 0; m < M_MICS; ++m)
        async_cplx_to_lds(X + ((size_t)m*T_FRAMES + (t+1))*F_BINS + f, nxt + m);
    }

    // y with carried W; per-bin energies -> global pooled r
    float e[K_SRC];
    #pragma unroll
    for (int k = 0; k < K_SRC; ++k){
      cx y = {0,0};
      #pragma unroll
      for (int j = 0; j < M_MICS; ++j) y = cmac(y, W[k][j], x[j]);
      e[k] = fmaf(y.r, y.r, y.i*y.i);
    }

    // Cross-lane energy reduction on the matrix pipe (fully-converged region,
    // EXEC all-1s as WMMA requires): A = [e0,e1] per lane (16x4 layout pairs
    // lane m with lane m+16 in K), B = column selectors ->
    // D[m][0] = e0(m)+e0(m+16), D[m][1] = e1(m)+e1(m+16).
    // Remaining fold: 7 adds over D's 8 VGPRs + one shfl_xor(16); both wave
    // totals land in lanes 0 (e0) and 1 (e1) simultaneously.
    {
      v2f A; A.x = e[0]; A.y = e[1];
      v8f Cz = {};
      v8f D = __builtin_amdgcn_wmma_f32_16x16x4_f32(false, A, false, Bsel,
                                                    (short)0, Cz, false, false);
      float s = ((D[0]+D[1]) + (D[2]+D[3])) + ((D[4]+D[5]) + (D[6]+D[7]));
      s += __shfl_xor(s, 16, 32);
      if (lane == 0) sred[0][wave] = s;
      if (lane == 1) sred[1][wave] = s;
    }
    __syncthreads();
    if (tid == 0){
      #pragma unroll
      for (int k = 0; k < K_SRC; ++k){
        float s = 0.f;
        #pragma unroll
        for (int w2 = 0; w2 < NWAVE; ++w2) s += sred[k][w2];
        __hip_atomic_store(&pbuf[blockIdx.x*K_SRC + k], s,
                           __ATOMIC_RELAXED, __HIP_MEMORY_SCOPE_AGENT);
      }
    }

    // single device-wide sync per step (ping-pong partials make the 2nd redundant)
    grid_barrier(cnt, gen, in_cluster);

    // leader computes phi (deterministic fixed-order sum), broadcasts via LDS
    if (tid == 0){
      #pragma unroll
      for (int k = 0; k < K_SRC; ++k){
        float s = 0.f;
        for (int g = 0; g < (int)gridDim.x; ++g)
          s += __hip_atomic_load(&pbuf[g*K_SRC + k], __ATOMIC_RELAXED, __HIP_MEMORY_SCOPE_AGENT);
        float r = sqrtf(fmaxf(s, EPS_R_F));
        sphi[k] = 1.0f / fmaxf(r, EPS_R_F);
      }
    }
    __syncthreads();
    float phi[K_SRC];
    #pragma unroll
    for (int k = 0; k < K_SRC; ++k) phi[k] = sphi[k];

    if (act){
      const float bC = 1.0f - ALPHA_F;
      // C = a*C + (1-a) x x^H  (b128 in / b128 out, L2-resident)
      {
        mat6 Cm; m_load(Cm, cp);
        #pragma unroll
        for (int i2 = 0; i2 < M_MICS; ++i2)
          #pragma unroll
          for (int j = 0; j < M_MICS; ++j){
            cx c0 = m_get(Cm, i2, j);
            float xr = fmaf(x[i2].r, x[j].r,  x[i2].i*x[j].i);
            float xi = fmaf(x[i2].i, x[j].r, -x[i2].r*x[j].i);
            m_set(Cm, i2, j, { fmaf(ALPHA_F, c0.r, bC*xr), fmaf(ALPHA_F, c0.i, bC*xi) });
          }
        m_store(Cm, cp);
      }

      // sequential IP update per source
      #pragma unroll
      for (int k = 0; k < K_SRC; ++k){
        float* vp = (k == 0) ? vp0 : vp1;
        const float bV = (1.0f - ALPHA_F) * phi[k];

        // V <- a*V + (1-a)*phi_k * x x^H ; keep (V + reg*I) in regs for the matmul
        mat6 Vm; m_load(Vm, vp);
        #pragma unroll
        for (int i2 = 0; i2 < M_MICS; ++i2)
          #pragma unroll
          for (int j = 0; j < M_MICS; ++j){
            cx v0 = m_get(Vm, i2, j);
            float xr = fmaf(x[i2].r, x[j].r,  x[i2].i*x[j].i);
            float xi = fmaf(x[i2].i, x[j].r, -x[i2].r*x[j].i);
            m_set(Vm, i2, j, { fmaf(ALPHA_F, v0.r, bV*xr), fmaf(ALPHA_F, v0.i, bV*xi) });
          }
        m_store(Vm, vp);
        #pragma unroll
        for (int d = 0; d < M_MICS; ++d){
          cx v0 = m_get(Vm, d, d); v0.r += REG_F; m_set(Vm, d, d, v0);
        }

        // temp = W @ (V + reg I)   (6x6 complex)
        cx tm[M_MICS][M_MICS];
        #pragma unroll
        for (int i2 = 0; i2 < M_MICS; ++i2)
          #pragma unroll
          for (int j = 0; j < M_MICS; ++j){
            cx a = {0,0};
            #pragma unroll
            for (int m = 0; m < M_MICS; ++m) a = cmac(a, W[i2][m], m_get(Vm, m, j));
            tm[i2][j] = a;
          }

        // solve tm * w = e_k (in-register Gaussian elimination)
        cx wv[M_MICS];
        #pragma unroll
        for (int i2 = 0; i2 < M_MICS; ++i2) wv[i2] = { (i2==k) ? 1.f : 0.f, 0.f };
        #pragma unroll
        for (int c = 0; c < M_MICS; ++c){
          cx inv = cinv(tm[c][c]);
          tm[c][c] = inv;
          #pragma unroll
          for (int r2 = c+1; r2 < M_MICS; ++r2){
            cx fac = cmul(tm[r2][c], inv);
            #pragma unroll
            for (int j = c+1; j < M_MICS; ++j) tm[r2][j] = cmsub(tm[r2][j], fac, tm[c][j]);
            wv[r2] = cmsub(wv[r2], fac, wv[c]);
          }
        }
        #pragma unroll
        for (int c = M_MICS-1; c >= 0; --c){
          cx s = wv[c];
          #pragma unroll
          for (int j = c+1; j < M_MICS; ++j) s = cmsub(s, tm[c][j], wv[j]);
          wv[c] = cmul(s, tm[c][c]);
        }

        // quad = real(w^H (V+regI) w); re-read V (b128, L0-hot) instead of
        // keeping a third 6x6 live across the solve (caps VGPR pressure)
        mat6 Vq; m_load(Vq, vp);
        float quad = 0.f;
        #pragma unroll
        for (int i2 = 0; i2 < M_MICS; ++i2){
          cx acc = {0,0};
          #pragma unroll
          for (int j = 0; j < M_MICS; ++j){
            cx vv = m_get(Vq, i2, j);
            if (i2 == j) vv.r += REG_F;
            acc = cmac(acc, vv, wv[j]);
          }
          quad = fmaf(wv[i2].r, acc.r, fmaf(wv[i2].i, acc.i, quad));
        }
        float sc = rsqrtf(quad + EPS_R_F);
        #pragma unroll
        for (int j = 0; j < M_MICS; ++j) W[k][j] = { wv[j].r*sc, -(wv[j].i*sc) };
      }

      // y_out depends only on W rows 0..K-1 (final after IP solves); store it
      // now (non-temporal: output is write-once, keep L0/L2 for C/V state) and
      // let the J-update overlap the store latency.
      #pragma unroll
      for (int k = 0; k < K_SRC; ++k){
        cx y = {0,0};
        #pragma unroll
        for (int j = 0; j < M_MICS; ++j) y = cmac(y, W[k][j], x[j]);
        v2f yo; yo.x = y.r; yo.y = y.i;
        __builtin_nontemporal_store(yo,
            (v2f*)(out + ((size_t)k*T_FRAMES + t)*F_BINS + f));
      }

      // background block: tmp = W[:K] @ C ; J = solve(tmp[:,:K], tmp[:,K:])^H
      // (writes only W rows K..M-1, consumed by next step's solves)
      cx tmp[K_SRC][M_MICS];
      #pragma unroll
      for (int i2 = 0; i2 < K_SRC; ++i2)
        #pragma unroll
        for (int j = 0; j < M_MICS; ++j) tmp[i2][j] = {0,0};
      {
        mat6 Cm; m_load(Cm, cp);                    // b128 re-read (L0-hot)
        #pragma unroll
        for (int m = 0; m < M_MICS; ++m)
          #pragma unroll
          for (int j = 0; j < M_MICS; ++j){
            cx cc = m_get(Cm, m, j);
            #pragma unroll
            for (int i2 = 0; i2 < K_SRC; ++i2) tmp[i2][j] = cmac(tmp[i2][j], W[i2][m], cc);
          }
      }
      cx A00 = tmp[0][0], A01 = tmp[0][1], A10 = tmp[1][0], A11 = tmp[1][1];
      cx idet = cinv(cmsub(cmul(A00, A11), A01, A10));
      #pragma unroll
      for (int c = 0; c < M_MICS - K_SRC; ++c){
        cx b0 = tmp[0][K_SRC+c], b1 = tmp[1][K_SRC+c];
        cx j0 = cmul(cmsub(cmul(A11, b0), A01, b1), idet);
        cx j1 = cmul(cmsub(cmul(A00, b1), A10, b0), idet);
        W[K_SRC+c][0] = { j0.r, -j0.i };
        W[K_SRC+c][1] = { j1.r, -j1.i };
      }
    }
    // no second barrier: partials are ping-ponged on t&1, and barrier(t+1)
    // already orders step-(t+2) writers after all step-t readers.
  }
}

extern "C" void kernel_launch(void* const* d_in, const int* in_sizes, int n_in,
                              void* d_out, int out_size, void* d_ws, size_t ws_size,
                              hipStream_t stream) {
  (void)in_sizes; (void)n_in; (void)out_size; (void)ws_size;
  const float2* X  = (const float2*)d_in[0];   // (M, T, F, 2) f32 == (M,T,F) cplx
  float2* out      = (float2*)d_out;           // (K, T, F, 2) f32
  float*  ws       = (float*)d_ws;             // needs >= (WS_V + 2*F_BINS*72)*4 B (~886 KB)

  hipLaunchKernelGGL(oiva_init, dim3((F_BINS + 255) / 256), dim3(256), 0, stream, ws);
  hipLaunchKernelGGL(oiva_persist, dim3(NWG), dim3(BLOCK), 0, stream, X, out, ws);
}